// LigerMultiTokenAttention_12919261626430
// MI455X (gfx1250) — compile-verified
//
#include <hip/hip_runtime.h>
#include <hip/hip_bf16.h>

// Problem constants (match reference).
#define L_DIM   1024
#define LL_DIM  (1024 * 1024)
#define B_DIM   4
#define H_DIM   16
#define TILES   (B_DIM * L_DIM * (L_DIM / 16))   // 262144 output tiles (16 px)
#define STRIPS  (TILES / 4)                      // 65536 wave-strips (4 tiles)

// Padded channel-last probs: probs_p[b][1026][1026][16], halo ring of zeros.
#define PR_DIM     1026
#define ROWSTRIDE  (PR_DIM * 16)                 // floats per padded row
#define PB_STRIDE  ((size_t)PR_DIM * ROWSTRIDE)  // floats per batch image

// Workspace layout (floats): [0..4096) transposed weights, [4096..) probs_p.
#define WT_OFF  0
#define PP_OFF  4096

typedef float v2f __attribute__((ext_vector_type(2)));
typedef float v8f __attribute__((ext_vector_type(8)));

// ---------------------------------------------------------------------------
// Kernel 0a: weight transpose OIHW -> [tap][oc][ic] (one b64 per A chunk).
// ---------------------------------------------------------------------------
__global__ __launch_bounds__(256) void mta_wtr_kernel(
    const float* __restrict__ w, float* __restrict__ wt) {
  const int idx = blockIdx.x * 256 + threadIdx.x;
  if (idx < H_DIM * H_DIM * 9) {
    const int tap = idx % 9;
    const int ic  = (idx / 9) % 16;
    const int oc  = idx / 144;
    wt[(tap * 16 + oc) * 16 + ic] = w[idx];
  }
}

// ---------------------------------------------------------------------------
// Kernel 0b: zero the top (pr=0) and bottom (pr=1025) halo rows of probs_p.
// 4 * 2 * 1026 * 16 = 131328 floats.
// ---------------------------------------------------------------------------
__global__ __launch_bounds__(256) void mta_zero_halo_kernel(float* __restrict__ pp) {
  const int idx = blockIdx.x * 256 + threadIdx.x;
  const int per_b = 2 * ROWSTRIDE;               // two halo rows per batch
  if (idx < B_DIM * per_b) {
    const int b    = idx / per_b;
    const int rem  = idx % per_b;
    const int side = rem / ROWSTRIDE;            // 0 = top, 1 = bottom
    const int off  = rem % ROWSTRIDE;
    const size_t pr = side ? (PR_DIM - 1) : 0;
    pp[(size_t)b * PB_STRIDE + pr * ROWSTRIDE + off] = 0.0f;
  }
}

// ---------------------------------------------------------------------------
// Kernel 1: causal softmax -> padded channel-last probs. One 512-thread
// block per (b, q); wave h handles head h so the 16 waves fill complementary
// 4B slices of the same cachelines (L2 merges). Writes exact zeros above the
// diagonal and in the left/right column halo.
// ---------------------------------------------------------------------------
__global__ __launch_bounds__(512) void mta_softmax_t_kernel(
    const float* __restrict__ scores, float* __restrict__ probs_p) {
  const int lane = threadIdx.x & 31;
  const int h    = threadIdx.x >> 5;             // 0..15 head per wave
  const int b    = blockIdx.x >> 10;
  const int q    = blockIdx.x & (L_DIM - 1);
  const int nval = q + 1;

  const float* src = scores + ((size_t)(b * H_DIM + h) * L_DIM + q) * L_DIM;
  // Row base at padded (pr = q+1, pc = 0), channel h.
  float* dst = probs_p + (size_t)b * PB_STRIDE +
               (size_t)(q + 1) * ROWSTRIDE + h;

  float vals[32];
  float m = -__builtin_inff();
#pragma unroll
  for (int i = 0; i < 32; ++i) {
    const int c = i * 32 + lane;
    const float x = (c < nval) ? src[c] : -__builtin_inff();
    vals[i] = x;
    m = fmaxf(m, x);
  }
#pragma unroll
  for (int off = 16; off > 0; off >>= 1) m = fmaxf(m, __shfl_xor(m, off, 32));

  float s = 0.0f;
#pragma unroll
  for (int i = 0; i < 32; ++i) {
    const int c = i * 32 + lane;
    const float e = (c < nval) ? __expf(vals[i] - m) : 0.0f;
    vals[i] = e;
    s += e;
  }
#pragma unroll
  for (int off = 16; off > 0; off >>= 1) s += __shfl_xor(s, off, 32);

  const float inv = 1.0f / s;
#pragma unroll
  for (int i = 0; i < 32; ++i)
    dst[(size_t)(1 + i * 32 + lane) * 16] = vals[i] * inv;   // pc = 1 + col

  // Left/right column halo for this row & head.
  if (lane == 0) dst[0] = 0.0f;                              // pc = 0
  if (lane == 1) dst[(size_t)(PR_DIM - 1) * 16] = 0.0f;      // pc = 1025
}

// ---------------------------------------------------------------------------
// Kernel 2: 3x3 conv (16ch->16ch) + bias + causal zero via f32 WMMA.
// One wave32 per 4-tile strip (64 px of one row q). A chunks live in 72
// VGPRs; every B load is anchor + compile-time immediate offset (padded
// halo removes all bounds logic). 144 WMMAs per wave.
// ---------------------------------------------------------------------------
__global__ __launch_bounds__(256) void mta_conv_wmma_kernel(
    const float* __restrict__ probs_p, const float* __restrict__ wt,
    const float* __restrict__ bias, float* __restrict__ out) {
  const int lane  = threadIdx.x & 31;
  const int gwave = blockIdx.x * 8 + (threadIdx.x >> 5);  // exact grid

  const int b      = gwave >> 14;        // 16384 strips per batch
  const int rem    = gwave & 16383;
  const int q      = rem >> 4;           // row
  const int k0base = (rem & 15) << 6;    // strip start col (64 px)

  const int n  = lane & 15;              // pixel / N index (both halves)
  const int oc = lane & 15;              // A-matrix M index (both halves)
  const int hi = (lane >= 16) ? 1 : 0;   // half-wave K-pair selector
  const int qu = __builtin_amdgcn_readfirstlane(q);

  // Preload all 36 A chunks: one anchor, constant offsets.
  const float* wa = wt + oc * 16 + hi * 2;
  v2f a[36];
#pragma unroll
  for (int tap = 0; tap < 9; ++tap) {
#pragma unroll
    for (int cb = 0; cb < 4; ++cb)
      a[tap * 4 + cb] = *(const v2f*)(wa + tap * 256 + cb * 4);
  }

  float bias_r[8];
#pragma unroll
  for (int v = 0; v < 8; ++v) bias_r[v] = bias[v + hi * 8];

  // B anchor: padded (pr = q + kh) rows, (pc = k0base + 16t + n + kw) cols.
  // For kh = 0, t = 0, kw = 0, cb = 0 -> anchor below; everything else is a
  // compile-time delta (max ~134 KB, inside signed 24-bit IOFFSET).
  const float* anchor = probs_p + (size_t)b * PB_STRIDE +
                        (size_t)q * ROWSTRIDE +
                        (size_t)(k0base + n) * 16 + hi * 2;

#pragma unroll
  for (int t = 0; t < 4; ++t) {
    const int k0 = k0base + (t << 4);

    v8f acc;
#pragma unroll
    for (int v = 0; v < 8; ++v) acc[v] = bias_r[v];

    // Uniform scalar branch: EXEC stays all-ones around every WMMA.
    if (__builtin_amdgcn_readfirstlane(k0) <= qu) {
#pragma unroll
      for (int kh = 0; kh < 3; ++kh) {
#pragma unroll
        for (int kw = 0; kw < 3; ++kw) {
#pragma unroll
          for (int cb = 0; cb < 4; ++cb) {
            const int off = (kh * PR_DIM + t * 16 + kw) * 16 + cb * 4;
            const v2f bv = *(const v2f*)(anchor + off);
            acc = __builtin_amdgcn_wmma_f32_16x16x4_f32(
                /*neg_a=*/false, a[(kh * 3 + kw) * 4 + cb], /*neg_b=*/false,
                bv, /*c_mod=*/(short)0, acc,
                /*reuse_a=*/false, /*reuse_b=*/false);
          }
        }
      }
    }

    // Store with causal zeroing (bias excluded above the diagonal).
    const int  colo = k0 + n;
    const bool keep = (colo <= q);
#pragma unroll
    for (int v = 0; v < 8; ++v) {
      const int oc_v = v + hi * 8;
      out[(((size_t)b * H_DIM + oc_v) * L_DIM + q) * L_DIM + colo] =
          keep ? acc[v] : 0.0f;
    }
  }
}

// ---------------------------------------------------------------------------
extern "C" void kernel_launch(void* const* d_in, const int* in_sizes, int n_in,
                              void* d_out, int out_size, void* d_ws, size_t ws_size,
                              hipStream_t stream) {
  const float* scores = (const float*)d_in[0];  // [4,16,1024,1024] f32
  const float* weight = (const float*)d_in[1];  // [16,16,3,3] f32 (OIHW)
  const float* bias   = (const float*)d_in[2];  // [16] f32
  float*       out    = (float*)d_out;          // [4,16,1024,1024] f32
  float*       ws     = (float*)d_ws;

  float* wt = ws + WT_OFF;                      // 2304 floats
  float* pp = ws + PP_OFF;                      // 4*1026*1026*16 floats

  (void)in_sizes; (void)n_in; (void)out_size; (void)ws_size;

  // 0a: weight transpose (2304 elems).
  mta_wtr_kernel<<<9, 256, 0, stream>>>(weight, wt);
  // 0b: zero top/bottom halo rows (131328 elems -> 513 blocks).
  mta_zero_halo_kernel<<<513, 256, 0, stream>>>(pp);
  // 1: softmax, 4096 (b,q) blocks x 512 threads (16 waves = 16 heads).
  mta_softmax_t_kernel<<<B_DIM * L_DIM, 512, 0, stream>>>(scores, pp);
  // 2: conv, 65536 strips, 8 waves/block -> 8192 blocks (exact).
  mta_conv_wmma_kernel<<<STRIPS / 8, 256, 0, stream>>>(pp, wt, bias, out);
}